// Sample_loss_44092134261091
// MI455X (gfx1250) — compile-verified
//
#include <hip/hip_runtime.h>
#include <hip/hip_bf16.h>

// Problem constants from the reference: N=16384, MAXL=2048, C=4096, B=N/8.
#define N_TOTAL 16384
#define MAXL    2048
#define CDIM    4096
#define BROWS   (N_TOTAL / 8)   // 2048 rows actually used
#define BLOCK   256             // 8 wave32s per workgroup

typedef __attribute__((address_space(3))) unsigned char lds_byte_t;

// --- Kernel 1: one workgroup per row b ---------------------------------------
// Stage x[b, 0:C] (16 KB) into LDS with CDNA5 async global->LDS b128 copies.
// While the async DMA is in flight, stream this thread's 8 y-indices with two
// b128 loads (independent LOADcnt vs ASYNCcnt). Then gather from LDS, mask by
// length, and tree-reduce the product.
__global__ __launch_bounds__(BLOCK) void row_prod_kernel(
    const float* __restrict__ x,
    const int*   __restrict__ y,
    const int*   __restrict__ lengths,
    float*       __restrict__ partial)
{
    __shared__ float xrow[CDIM];     // 16 KB staged row of x
    __shared__ float red[BLOCK];

    const int b   = blockIdx.x;
    const int tid = threadIdx.x;

    // ---- async stage: 4 rounds x 256 lanes x 16B = 16 KB -------------------
    const float* xg = x + (size_t)b * CDIM;
    unsigned lds_base = (unsigned)(size_t)(lds_byte_t*)xrow;  // LDS byte offset

    #pragma unroll
    for (int r = 0; r < 4; ++r) {
        unsigned     lds_addr = lds_base + (unsigned)(r * 4096 + tid * 16);
        const float* gaddr    = xg + r * 1024 + tid * 4;
        asm volatile("global_load_async_to_lds_b128 %0, %1, off"
                     :: "v"(lds_addr), "v"(gaddr)
                     : "memory");
    }

    // ---- overlap: y row always has MAXL valid entries; 2048/4 = 2*BLOCK ----
    // Each thread owns chunks l = 4*tid+{0..3} and l = 4*(tid+BLOCK)+{0..3}.
    const int4* yrow4 = (const int4*)(y + (size_t)b * MAXL);
    int4 ya = yrow4[tid];            // global_load_b128, concurrent with async
    int4 yb = yrow4[tid + BLOCK];    // global_load_b128
    const int len = lengths[b];      // scalar load (b is uniform)

    asm volatile("s_wait_asynccnt 0" ::: "memory");  // drain async LDS fill
    __syncthreads();                                  // all waves' copies visible

    // ---- 8 LDS gathers, length-masked product ------------------------------
    const int l0 = 4 * tid;
    const int l1 = 4 * (tid + BLOCK);

    float p = 1.0f;
    p *= (l0 + 0 < len) ? (1.0f - xrow[ya.x]) : 1.0f;
    p *= (l0 + 1 < len) ? (1.0f - xrow[ya.y]) : 1.0f;
    p *= (l0 + 2 < len) ? (1.0f - xrow[ya.z]) : 1.0f;
    p *= (l0 + 3 < len) ? (1.0f - xrow[ya.w]) : 1.0f;
    p *= (l1 + 0 < len) ? (1.0f - xrow[yb.x]) : 1.0f;
    p *= (l1 + 1 < len) ? (1.0f - xrow[yb.y]) : 1.0f;
    p *= (l1 + 2 < len) ? (1.0f - xrow[yb.z]) : 1.0f;
    p *= (l1 + 3 < len) ? (1.0f - xrow[yb.w]) : 1.0f;

    // ---- product reduction across the 256 threads --------------------------
    red[tid] = p;
    __syncthreads();
    #pragma unroll
    for (int s = BLOCK / 2; s > 0; s >>= 1) {
        if (tid < s) red[tid] *= red[tid + s];
        __syncthreads();
    }
    if (tid == 0) partial[b] = 1.0f - red[0];
}

// --- Kernel 2: deterministic single-block sum of the 2048 per-row values ----
__global__ __launch_bounds__(BLOCK) void reduce_kernel(
    const float* __restrict__ partial,
    float*       __restrict__ out)
{
    __shared__ float red[BLOCK];
    float s = 0.0f;
    for (int i = threadIdx.x; i < BROWS; i += BLOCK) s += partial[i];
    red[threadIdx.x] = s;
    __syncthreads();
    #pragma unroll
    for (int k = BLOCK / 2; k > 0; k >>= 1) {
        if (threadIdx.x < k) red[threadIdx.x] += red[threadIdx.x + k];
        __syncthreads();
    }
    if (threadIdx.x == 0) out[0] = red[0];
}

extern "C" void kernel_launch(void* const* d_in, const int* in_sizes, int n_in,
                              void* d_out, int out_size, void* d_ws, size_t ws_size,
                              hipStream_t stream) {
    (void)in_sizes; (void)n_in; (void)out_size; (void)ws_size;
    const float* x       = (const float*)d_in[0];
    const int*   y       = (const int*)d_in[1];
    const int*   lengths = (const int*)d_in[2];
    float*       out     = (float*)d_out;
    float*       ws      = (float*)d_ws;   // BROWS floats = 8 KB of scratch

    row_prod_kernel<<<BROWS, BLOCK, 0, stream>>>(x, y, lengths, ws);
    reduce_kernel<<<1, BLOCK, 0, stream>>>(ws, out);
}